// InvoiceGCN_28956669510215
// MI455X (gfx1250) — compile-verified
//
#include <hip/hip_runtime.h>
#include <hip/hip_bf16.h>
#include <cstddef>

#define N_NODES 10000
#define N_EDGES 160000
#define F1_REAL 782
#define F1_PAD  784   // padded to multiple of 4 for WMMA K-steps

typedef __attribute__((ext_vector_type(2))) float v2f;
typedef __attribute__((ext_vector_type(8))) float v8f;

// ---------------- graph norm ----------------

__global__ __launch_bounds__(256)
void deg_kernel(const int* __restrict__ src, float* __restrict__ deg, int E) {
    int e = blockIdx.x * blockDim.x + threadIdx.x;
    if (e < E) atomicAdd(&deg[src[e]], 1.0f);
}

__global__ __launch_bounds__(256)
void dinv_kernel(float* __restrict__ deg, int n) {
    int i = blockIdx.x * blockDim.x + threadIdx.x;
    if (i < n) {
        float d = deg[i];
        deg[i] = (d > 0.0f) ? (1.0f / sqrtf(d)) : 0.0f;
    }
}

__global__ __launch_bounds__(256)
void norm_kernel(const int* __restrict__ src, const int* __restrict__ dst,
                 const float* __restrict__ dinv, float* __restrict__ norm, int E) {
    int e = blockIdx.x * blockDim.x + threadIdx.x;
    if (e < E) norm[e] = -dinv[src[e]] * dinv[dst[e]];
}

// ---------------- input pad: xpad[r, 0:782] = x, xpad[r, 782:784] = 0 -------

__global__ __launch_bounds__(256)
void xpad_kernel(const float* __restrict__ x, float* __restrict__ xp, size_t n) {
    size_t i = (size_t)blockIdx.x * blockDim.x + threadIdx.x;
    if (i >= n) return;
    int r = (int)(i / F1_PAD), k = (int)(i % F1_PAD);
    xp[i] = (k < F1_REAL) ? x[(size_t)r * F1_REAL + k] : 0.0f;
}

// ------------- weight prep: Wt[k][n][ki] = W[k][ki][n], zero-padded ---------

__global__ __launch_bounds__(256)
void wprep_kernel(const float* __restrict__ W, float* __restrict__ Wt,
                  int Fin, int Fout, int Finp, int Foutp) {
    int total = 3 * Foutp * Finp;
    int i = blockIdx.x * blockDim.x + threadIdx.x;
    if (i >= total) return;
    int k  = i / (Foutp * Finp);
    int r  = i % (Foutp * Finp);
    int n  = r / Finp;
    int ki = r % Finp;
    Wt[i] = (ki < Fin && n < Fout) ? W[((size_t)k * Fin + ki) * Fout + n] : 0.0f;
}

__global__ void bprep_kernel(const float* __restrict__ b, float* __restrict__ bp,
                             int Fout, int Foutp) {
    int i = threadIdx.x;
    if (i < Foutp) bp[i] = (i < Fout) ? b[i] : 0.0f;
}

// ---------------- SpMM: y[dst[e],:] += norm[e] * x[src[e],:] ----------------

__global__ __launch_bounds__(128)
void spmm_kernel(const float* __restrict__ norm, const int* __restrict__ src,
                 const int* __restrict__ dst, const float* __restrict__ x,
                 float* __restrict__ y, int F) {
    int e = blockIdx.x;
    float w = norm[e];
    const float* xs = x + (size_t)src[e] * F;
    float*       yd = y + (size_t)dst[e] * F;
    for (int f = threadIdx.x; f < F; f += blockDim.x)
        atomicAdd(&yd[f], w * xs[f]);
}

// ---------------- T2 = 2*T2 - T0 (Chebyshev recursion) ----------------------

__global__ __launch_bounds__(256)
void cheb2_kernel(float* __restrict__ T2, const float* __restrict__ T0, size_t n) {
    size_t i = (size_t)blockIdx.x * blockDim.x + threadIdx.x;
    if (i < n) T2[i] = 2.0f * T2[i] - T0[i];
}

// ---------------- WMMA GEMM: out = T0@W0 + T1@W1 + T2@W2 + b (+relu) --------
// fp32 V_WMMA_F32_16X16X4_F32. One wave owns a 16-row stripe and ALL TN
// column tiles (register blocking): each A float2 is reused for TN WMMAs,
// cutting loads-per-WMMA from 2 to 1 + 1/TN and L2 A-traffic by TN x.
// A 16x4 fragment: lane(h=lane>>4,lid=lane&15) -> row=lid, VGPR r holds K=2h+r.
// B 4x16 fragment mirrors it (Wt transposed -> contiguous float2).
// C/D: VGPR r holds (M=r+8h, N=lid).

template<int FINP, int TN>
__global__ __launch_bounds__(32)
void cheb_gemm_wmma(const float* __restrict__ T0, const float* __restrict__ T1,
                    const float* __restrict__ T2, const float* __restrict__ Wt,
                    const float* __restrict__ bias, float* __restrict__ out,
                    int relu) {
    constexpr int FOUTP = TN * 16;
    int tm   = blockIdx.x;
    int lane = threadIdx.x;
    int half = lane >> 4;
    int lid  = lane & 15;
    int row  = tm * 16 + lid;
    int base = 2 * half;

    v8f c[TN];
#pragma unroll
    for (int j = 0; j < TN; ++j) c[j] = (v8f){};

#pragma unroll
    for (int t = 0; t < 3; ++t) {
        const float* At = (t == 0) ? T0 : (t == 1) ? T1 : T2;
        const float* Ar = At + (size_t)row * FINP + base;
        const float* Bt = Wt + ((size_t)t * FOUTP + lid) * FINP + base;
#pragma unroll 4
        for (int k0 = 0; k0 < FINP; k0 += 4) {
            if ((k0 & 31) == 0) {
                __builtin_prefetch(Ar + k0 + 64, 0, 1);
            }
            v2f a = *(const v2f*)(Ar + k0);
#pragma unroll
            for (int j = 0; j < TN; ++j) {
                v2f b = *(const v2f*)(Bt + (size_t)j * 16 * FINP + k0);
                c[j] = __builtin_amdgcn_wmma_f32_16x16x4_f32(
                           false, a, false, b, (short)0, c[j], false, false);
            }
        }
    }

#pragma unroll
    for (int j = 0; j < TN; ++j) {
        int col = j * 16 + lid;
        float bv = bias[col];
#pragma unroll
        for (int r = 0; r < 8; ++r) {
            int orow = tm * 16 + r + 8 * half;
            float v = c[j][r] + bv;
            if (relu) v = fmaxf(v, 0.0f);
            out[(size_t)orow * FOUTP + col] = v;
        }
    }
}

// ---------------- row softmax (stride-32 logits -> dense N x 19) ------------

__global__ __launch_bounds__(256)
void softmax_kernel(const float* __restrict__ in, float* __restrict__ out,
                    int n, int F, int strideIn) {
    int i = blockIdx.x * blockDim.x + threadIdx.x;
    if (i >= n) return;
    const float* r = in + (size_t)i * strideIn;
    float m = r[0];
    for (int f = 1; f < F; ++f) m = fmaxf(m, r[f]);
    float s = 0.0f;
    for (int f = 0; f < F; ++f) s += expf(r[f] - m);
    float inv = 1.0f / s;
    for (int f = 0; f < F; ++f) out[(size_t)i * F + f] = expf(r[f] - m) * inv;
}

// ---------------- host driver ----------------

static inline size_t align256(size_t x) { return (x + 255) & ~(size_t)255; }

extern "C" void kernel_launch(void* const* d_in, const int* in_sizes, int n_in,
                              void* d_out, int out_size, void* d_ws, size_t ws_size,
                              hipStream_t stream) {
    (void)in_sizes; (void)n_in; (void)out_size; (void)ws_size;

    const float* x  = (const float*)d_in[0];
    const int*   ei = (const int*)d_in[1];
    const float* W[5]; const float* B[5];
    for (int i = 0; i < 5; ++i) {
        W[i] = (const float*)d_in[2 + 2 * i];
        B[i] = (const float*)d_in[3 + 2 * i];
    }
    const int* src = ei;
    const int* dst = ei + N_EDGES;

    const int FinR[5]  = {F1_REAL, 16, 32, 64, 128};
    const int Finp[5]  = {F1_PAD,  16, 32, 64, 128};
    const int FoutR[5] = {16, 32, 64, 128, 19};
    const int Foutp[5] = {16, 32, 64, 128, 32};

    // workspace carve-up
    char* p = (char*)d_ws;
    float* deg    = (float*)p; p += align256((size_t)N_NODES * 4);
    float* norm   = (float*)p; p += align256((size_t)N_EDGES * 4);
    float* xpad   = (float*)p; p += align256((size_t)N_NODES * F1_PAD * 4);
    float* T1     = (float*)p; p += align256((size_t)N_NODES * F1_PAD * 4);
    float* T2     = (float*)p; p += align256((size_t)N_NODES * F1_PAD * 4);
    float* hA     = (float*)p; p += align256((size_t)N_NODES * 128 * 4);
    float* hB     = (float*)p; p += align256((size_t)N_NODES * 128 * 4);
    float* logits = (float*)p; p += align256((size_t)N_NODES * 32 * 4);
    float* Wt[5]; float* bp[5];
    for (int L = 0; L < 5; ++L) {
        Wt[L] = (float*)p; p += align256((size_t)3 * Foutp[L] * Finp[L] * 4);
        bp[L] = (float*)p; p += align256((size_t)Foutp[L] * 4);
    }

    // ----- prep: norm, padded input, transposed+padded weights -----
    hipMemsetAsync(deg, 0, (size_t)N_NODES * 4, stream);
    deg_kernel<<<(N_EDGES + 255) / 256, 256, 0, stream>>>(src, deg, N_EDGES);
    dinv_kernel<<<(N_NODES + 255) / 256, 256, 0, stream>>>(deg, N_NODES);
    norm_kernel<<<(N_EDGES + 255) / 256, 256, 0, stream>>>(src, dst, deg, norm, N_EDGES);

    {
        size_t n = (size_t)N_NODES * F1_PAD;
        xpad_kernel<<<(int)((n + 255) / 256), 256, 0, stream>>>(x, xpad, n);
    }
    for (int L = 0; L < 5; ++L) {
        int total = 3 * Foutp[L] * Finp[L];
        wprep_kernel<<<(total + 255) / 256, 256, 0, stream>>>(
            W[L], Wt[L], FinR[L], FoutR[L], Finp[L], Foutp[L]);
        bprep_kernel<<<1, 128, 0, stream>>>(B[L], bp[L], FoutR[L], Foutp[L]);
    }

    // ----- layers -----
    const float* T0ptr[5] = {xpad, hA, hB, hA, hB};
    float*       Optr[5]  = {hA, hB, hA, hB, logits};
    const int grid = N_NODES / 16;   // 625 row stripes, exact

    for (int L = 0; L < 5; ++L) {
        int fi = Finp[L];
        size_t nfi = (size_t)N_NODES * fi;
        const float* T0 = T0ptr[L];
        float* O = Optr[L];
        int relu = (L < 4) ? 1 : 0;

        // T1 = Lhat @ T0 ; T2 = 2*(Lhat @ T1) - T0   (pad cols stay zero)
        hipMemsetAsync(T1, 0, nfi * 4, stream);
        spmm_kernel<<<N_EDGES, 128, 0, stream>>>(norm, src, dst, T0, T1, fi);
        hipMemsetAsync(T2, 0, nfi * 4, stream);
        spmm_kernel<<<N_EDGES, 128, 0, stream>>>(norm, src, dst, T1, T2, fi);
        cheb2_kernel<<<(int)((nfi + 255) / 256), 256, 0, stream>>>(T2, T0, nfi);

        switch (L) {
        case 0: cheb_gemm_wmma<F1_PAD, 1><<<grid, 32, 0, stream>>>(T0, T1, T2, Wt[L], bp[L], O, relu); break;
        case 1: cheb_gemm_wmma<16,     2><<<grid, 32, 0, stream>>>(T0, T1, T2, Wt[L], bp[L], O, relu); break;
        case 2: cheb_gemm_wmma<32,     4><<<grid, 32, 0, stream>>>(T0, T1, T2, Wt[L], bp[L], O, relu); break;
        case 3: cheb_gemm_wmma<64,     8><<<grid, 32, 0, stream>>>(T0, T1, T2, Wt[L], bp[L], O, relu); break;
        case 4: cheb_gemm_wmma<128,    2><<<grid, 32, 0, stream>>>(T0, T1, T2, Wt[L], bp[L], O, relu); break;
        }
    }

    // ----- softmax -> d_out -----
    softmax_kernel<<<(N_NODES + 255) / 256, 256, 0, stream>>>(
        logits, (float*)d_out, N_NODES, FoutR[4], Foutp[4]);
}